// RetrievalGate_50972671868992
// MI455X (gfx1250) — compile-verified
//
#include <hip/hip_runtime.h>
#include <hip/hip_bf16.h>

typedef __attribute__((ext_vector_type(2))) float v2f;
typedef __attribute__((ext_vector_type(8))) float v8f;

#define C_DIM 2048
#define RD    32
#define T_DIM 4096
#define NC    4096
#define B_DIM 4
#define TOPB  8

// ---------------------------------------------------------------------------
// Kernel 1: q = l2norm(qh @ W^T + b)   [16384 x 32]
// One wave computes one 16-row tile via V_WMMA_F32_16X16X4_F32, K=2048.
// W is staged into LDS as K-pairs, parity-split (even kp / odd kp arrays with
// a 128 B shim) so the two half-waves read disjoint bank halves:
// one ds_load_b64 per B operand, conflict-free. Staging itself uses the CDNA5
// async global->LDS path (ASYNCcnt).
// ---------------------------------------------------------------------------
__global__ __launch_bounds__(128) void proj_kernel(const float* __restrict__ qh,
                                                   const float* __restrict__ W,
                                                   const float* __restrict__ bias,
                                                   float* __restrict__ q_ws) {
    extern __shared__ v2f wpair[];
    v2f* wE = wpair;                      // even kp: [512][32]
    v2f* wO = wpair + 512 * RD + 16;      // odd kp, +128 B shim -> bank offset 32

    const int tid = threadIdx.x;

    // async scatter-stage: pair kp of W row n -> parity array, 8 B per copy
    for (int idx2 = tid; idx2 < RD * (C_DIM / 2); idx2 += 128) {
        const int n  = idx2 >> 10;        // / (C_DIM/2)
        const int kp = idx2 & 1023;
        v2f* dst = ((kp & 1) ? wO : wE) + (kp >> 1) * RD + n;
        const float* src = W + (size_t)n * C_DIM + 2 * kp;
        const unsigned lds_off = (unsigned)(uintptr_t)dst;
        asm volatile("global_load_async_to_lds_b64 %0, %1, off"
                     :: "v"(lds_off), "v"(src) : "memory");
    }
    asm volatile("s_wait_asynccnt 0x0" ::: "memory");
    __syncthreads();

    const int wave = tid >> 5;
    const int lane = tid & 31;
    const int n    = lane & 15;   // column within 16-wide tile / A row id
    const int h    = lane >> 4;   // half-wave selector (K pair 0..1 vs 2..3)

    const int t0 = (blockIdx.x * 4 + wave) * 16;
    const float* aptr = qh + (size_t)(t0 + n) * C_DIM + 2 * h;
    const v2f* bBase = h ? wO : wE;   // half h reads kp = k/2 + h -> row k/4

    v8f acc0 = {};  // columns n (0..15)
    v8f acc1 = {};  // columns 16+n

    for (int k0 = 0; k0 < C_DIM; k0 += 64) {
        __builtin_prefetch(aptr + k0 + 256, 0, 0);  // global_prefetch_b8
#pragma unroll
        for (int kk = 0; kk < 64; kk += 4) {
            const int k = k0 + kk;
            // A 16x4 fp32 layout: VGPR0 = K(2h), VGPR1 = K(2h+1)
            const v2f a  = *(const v2f*)(aptr + k);
            const v2f b0 = bBase[(k >> 2) * RD + n];
            const v2f b1 = bBase[(k >> 2) * RD + 16 + n];
            acc0 = __builtin_amdgcn_wmma_f32_16x16x4_f32(false, a, false, b0,
                                                         (short)0, acc0, false, false);
            acc1 = __builtin_amdgcn_wmma_f32_16x16x4_f32(false, a, false, b1,
                                                         (short)0, acc1, false, false);
        }
    }

    const float bn0 = bias[n];
    const float bn1 = bias[16 + n];
#pragma unroll
    for (int v = 0; v < 8; ++v) { acc0[v] += bn0; acc1[v] += bn1; }

    // row L2 norm: row (t0+v+8h) lives across the 16 lanes of this half-wave
#pragma unroll
    for (int v = 0; v < 8; ++v) {
        float s = acc0[v] * acc0[v] + acc1[v] * acc1[v];
        s += __shfl_xor(s, 1, 32);
        s += __shfl_xor(s, 2, 32);
        s += __shfl_xor(s, 4, 32);
        s += __shfl_xor(s, 8, 32);
        const float scale = 1.0f / fmaxf(sqrtf(s), 1e-12f);
        acc0[v] *= scale;
        acc1[v] *= scale;
    }

#pragma unroll
    for (int v = 0; v < 8; ++v) {
        const size_t row = (size_t)(t0 + v + 8 * h);
        q_ws[row * RD + n]      = acc0[v];
        q_ws[row * RD + 16 + n] = acc1[v];
    }
}

// ---------------------------------------------------------------------------
// Kernel 2: l2-normalize routing embeds (16384 rows of 32 floats, 2 MB)
// ---------------------------------------------------------------------------
__global__ __launch_bounds__(256) void rnorm_kernel(const float* __restrict__ r,
                                                    float* __restrict__ r_ws,
                                                    int nrows) {
    const int row = blockIdx.x * blockDim.x + threadIdx.x;
    if (row >= nrows) return;
    const float4* p = (const float4*)(r + (size_t)row * RD);
    float4 v[8];
    float s = 0.0f;
#pragma unroll
    for (int i = 0; i < 8; ++i) {
        v[i] = p[i];
        s += v[i].x * v[i].x + v[i].y * v[i].y + v[i].z * v[i].z + v[i].w * v[i].w;
    }
    const float scale = 1.0f / fmaxf(sqrtf(s), 1e-12f);
    float4* o = (float4*)(r_ws + (size_t)row * RD);
#pragma unroll
    for (int i = 0; i < 8; ++i) {
        float4 t = v[i];
        t.x *= scale; t.y *= scale; t.z *= scale; t.w *= scale;
        o[i] = t;
    }
}

// ---------------------------------------------------------------------------
// Kernel 3 (fused): scores for a 16-row strip (WMMA, K=32) written to global
// AND to a 256 KB LDS strip (row-padded +4 floats for bank-conflict freedom),
// then per-row top-8 straight from LDS -- saves the 256 MB scores re-read.
// ---------------------------------------------------------------------------
#define ROWP (NC + 4)

__global__ __launch_bounds__(256) void scores_topk_kernel(const float* __restrict__ q_ws,
                                                          const float* __restrict__ r_ws,
                                                          const unsigned char* __restrict__ mask,
                                                          float* __restrict__ scores,
                                                          int* __restrict__ out_idx) {
    extern __shared__ float tile[];   // [16][ROWP]
    const int tid  = threadIdx.x;
    const int wave = tid >> 5;
    const int lane = tid & 31;
    const int n    = lane & 15;
    const int h    = lane >> 4;

    const int b  = blockIdx.x >> 8;    // /256
    const int t0 = (blockIdx.x & 255) * 16;

    // A tile (16 rows x K=32) held in registers, reused across all 32 N-tiles
    const size_t qrow = (size_t)(b * T_DIM + t0 + n);
    v2f a[8];
#pragma unroll
    for (int ks = 0; ks < 8; ++ks)
        a[ks] = *(const v2f*)(q_ws + qrow * RD + 4 * ks + 2 * h);

    const float* rb = r_ws + (size_t)b * NC * RD;
    const unsigned char* mb = mask + (size_t)b * NC;

    for (int nt = wave; nt < NC / 16; nt += 8) {
        const int n0 = nt * 16;
        v8f acc = {};
#pragma unroll
        for (int ks = 0; ks < 8; ++ks) {
            const v2f bm = *(const v2f*)(rb + (size_t)(n0 + n) * RD + 4 * ks + 2 * h);
            acc = __builtin_amdgcn_wmma_f32_16x16x4_f32(false, a[ks], false, bm,
                                                        (short)0, acc, false, false);
        }
        const bool mok = mb[n0 + n] != 0;
#pragma unroll
        for (int v = 0; v < 8; ++v) {
            const float val = mok ? acc[v] : -__builtin_inff();
            const int   r   = v + 8 * h;                    // local row
            tile[r * ROWP + n0 + n] = val;
            scores[(size_t)(b * T_DIM + t0 + r) * NC + n0 + n] = val;
        }
    }
    __syncthreads();

    // top-8 per row from LDS: wave handles two local rows
#pragma unroll 1
    for (int rr = 0; rr < 2; ++rr) {
        const int r = wave * 2 + rr;
        const float* srow = tile + r * ROWP;

        float val[TOPB];
        int   idx[TOPB];
#pragma unroll
        for (int i = 0; i < TOPB; ++i) { val[i] = -__builtin_inff(); idx[i] = 0x7fffffff; }

        for (int j = lane; j < NC; j += 32) {
            const float v = srow[j];
            if (v > val[TOPB - 1]) {
                float cv = v; int ci = j;
#pragma unroll
                for (int i = 0; i < TOPB; ++i) {
                    if (cv > val[i]) {
                        const float tv = val[i]; const int ti = idx[i];
                        val[i] = cv; idx[i] = ci;
                        cv = tv; ci = ti;
                    }
                }
            }
        }

#pragma unroll 1
        for (int round = 0; round < TOPB; ++round) {
            float bv = val[0]; int bi = idx[0];
#pragma unroll
            for (int off = 16; off > 0; off >>= 1) {
                const float ov = __shfl_xor(bv, off, 32);
                const int   oi = __shfl_xor(bi, off, 32);
                if (ov > bv || (ov == bv && oi < bi)) { bv = ov; bi = oi; }
            }
            if (lane == 0)
                out_idx[(size_t)(b * T_DIM + t0 + r) * TOPB + round] = bi;
            if (idx[0] == bi && val[0] == bv) {  // owner pops its head
#pragma unroll
                for (int i = 0; i < TOPB - 1; ++i) { val[i] = val[i + 1]; idx[i] = idx[i + 1]; }
                val[TOPB - 1] = -__builtin_inff();
                idx[TOPB - 1] = 0x7fffffff;
            }
        }
    }
}

// ---------------------------------------------------------------------------
extern "C" void kernel_launch(void* const* d_in, const int* in_sizes, int n_in,
                              void* d_out, int out_size, void* d_ws, size_t ws_size,
                              hipStream_t stream) {
    (void)in_sizes; (void)n_in; (void)out_size; (void)ws_size;

    const float*         qh   = (const float*)d_in[0];          // [B,T,C]
    const float*         re   = (const float*)d_in[1];          // [B,NC,RD]
    const unsigned char* mask = (const unsigned char*)d_in[2];  // [B,NC] bool
    const float*         W    = (const float*)d_in[3];          // [RD,C]
    const float*         bias = (const float*)d_in[4];          // [RD]

    int*   idx_out    = (int*)d_out;                                  // [B,T,8]
    float* scores_out = (float*)d_out + (size_t)B_DIM * T_DIM * TOPB; // [B,T,NC]

    float* q_ws = (float*)d_ws;                               // 2 MB
    float* r_ws = q_ws + (size_t)B_DIM * T_DIM * RD;          // 2 MB

    // 1024 row-tiles of 16, 4 waves/block; (512*32 + 16 + 512*32) v2f LDS
    proj_kernel<<<256, 128, (2 * 512 * RD + 16) * sizeof(v2f), stream>>>(qh, W, bias, q_ws);
    rnorm_kernel<<<(B_DIM * NC + 255) / 256, 256, 0, stream>>>(re, r_ws, B_DIM * NC);
    scores_topk_kernel<<<B_DIM * (T_DIM / 16), 256, 16 * ROWP * sizeof(float), stream>>>(
        q_ws, r_ws, mask, scores_out, idx_out);
}